// ProcessNeurons_52201032515629
// MI455X (gfx1250) — compile-verified
//
#include <hip/hip_runtime.h>
#include <math.h>

typedef __attribute__((ext_vector_type(2))) float v2f;
typedef __attribute__((ext_vector_type(8))) float v8f;

#define B_   8
#define S_   2048
#define KIN_ 128
#define DM_  1024
#define NIN_ 256
#define NP_  1024
#define KP_  256

// per-wave macro-tile: 32 rows (2 x 16 M-subtiles) x 64 cols (4 x 16 N-subtiles)
#define MT_  2
#define NT_  4
#define TILE_M (MT_ * 16)   // 32
#define TILE_N (NT_ * 16)   // 64

// ---------------------------------------------------------------------------
// zero the router-score accumulator (B_*NP_ floats)
// ---------------------------------------------------------------------------
__global__ __launch_bounds__(256) void k_zero(float* __restrict__ scores) {
    int i = blockIdx.x * blockDim.x + threadIdx.x;
    if (i < B_ * NP_) scores[i] = 0.0f;
}

// ---------------------------------------------------------------------------
// GEMM1 + exact-erf GELU + column-sum accumulation for the router.
// One wave32 per 32x64 output macro-tile; 8 f32 WMMA accumulators.
//   acts[b,s,p] = gelu( sum_k A[b,s,k] * PW[p, IDX[b,k]] )
//   scores[b,p] += sum_s acts[b,s,p]   (sum == mean up to scale: top-k invariant)
// ---------------------------------------------------------------------------
__global__ __launch_bounds__(256) void k_gemm1(const float* __restrict__ A,
                                               const float* __restrict__ PW,
                                               const int*   __restrict__ IDX,
                                               float*       __restrict__ acts,
                                               float*       __restrict__ scores) {
    const int lane = threadIdx.x & 31;
    const int wave = (blockIdx.x * blockDim.x + threadIdx.x) >> 5;
    const int ntilesN = NP_ / TILE_N;                       // 16
    const int tilesPerBatch = (S_ / TILE_M) * ntilesN;      // 1024
    const int b  = wave / tilesPerBatch;
    const int t  = wave % tilesPerBatch;
    const int m0 = (t / ntilesN) * TILE_M;
    const int n0 = (t % ntilesN) * TILE_N;

    const int mn = lane & 15;          // M for A-frag, N for B-frag (ISA layout)
    const int kb = (lane >> 4) * 2;    // K sub-offset: VGPR0 holds K {0|2}, VGPR1 {1|3}

    const float* Arow0 = A + (size_t)(b * S_ + m0 + mn) * KIN_;        // M-subtile 0
    const float* Arow1 = Arow0 + (size_t)16 * KIN_;                    // M-subtile 1
    const float* PWr[NT_];
#pragma unroll
    for (int j = 0; j < NT_; ++j)
        PWr[j] = PW + (size_t)(n0 + j * 16 + mn) * NIN_;               // row p = n0+16j+n
    const int* idxb = IDX + b * KIN_;

    v8f c[MT_ * NT_];
#pragma unroll
    for (int i = 0; i < MT_ * NT_; ++i)
        c[i] = (v8f){0.f, 0.f, 0.f, 0.f, 0.f, 0.f, 0.f, 0.f};

#pragma unroll 2
    for (int k0 = 0; k0 < KIN_; k0 += 4) {
        v2f a0, a1;
        a0.x = Arow0[k0 + kb];  a0.y = Arow0[k0 + kb + 1];
        a1.x = Arow1[k0 + kb];  a1.y = Arow1[k0 + kb + 1];
        const int i0 = idxb[k0 + kb];
        const int i1 = idxb[k0 + kb + 1];
#pragma unroll
        for (int j = 0; j < NT_; ++j) {
            v2f bm;
            bm.x = PWr[j][i0];          // PW[p, idx[b,k]] gather (L2-resident)
            bm.y = PWr[j][i1];
            c[j * MT_ + 0] = __builtin_amdgcn_wmma_f32_16x16x4_f32(
                false, a0, false, bm, (short)0, c[j * MT_ + 0], false, false);
            c[j * MT_ + 1] = __builtin_amdgcn_wmma_f32_16x16x4_f32(
                false, a1, false, bm, (short)0, c[j * MT_ + 1], false, false);
        }
    }

    // C/D layout: VGPR r -> row r (lanes 0-15) / row 8+r (lanes 16-31), col = lane&15
    const int rbase = (lane >> 4) * 8;
    float csum[NT_] = {0.f, 0.f, 0.f, 0.f};
#pragma unroll
    for (int j = 0; j < NT_; ++j) {
#pragma unroll
        for (int mi = 0; mi < MT_; ++mi) {
            float* Crow = acts + (size_t)(b * S_ + m0 + mi * 16) * NP_ + n0 + j * 16 + mn;
            v8f cc = c[j * MT_ + mi];
#pragma unroll
            for (int r = 0; r < 8; ++r) {
                float x = cc[r];
                float g = 0.5f * x * (1.0f + erff(x * 0.70710678118654752f)); // exact gelu
                Crow[(size_t)(rbase + r) * NP_] = g;
                csum[j] += g;
            }
        }
    }
#pragma unroll
    for (int j = 0; j < NT_; ++j) {
        float v = csum[j] + __shfl_xor(csum[j], 16);  // combine 8-row halves of column
        if (lane < 16)
            atomicAdd(&scores[b * NP_ + n0 + j * 16 + lane], v);
    }
}

// ---------------------------------------------------------------------------
// Top-256-of-1024 per batch by rank counting (deterministic tie-break).
// ---------------------------------------------------------------------------
__global__ __launch_bounds__(256) void k_topk(const float* __restrict__ scores,
                                              int*         __restrict__ topk) {
    __shared__ float s[NP_];
    const int b = blockIdx.x;
    for (int i = threadIdx.x; i < NP_; i += 256) s[i] = scores[b * NP_ + i];
    __syncthreads();
    for (int p = threadIdx.x; p < NP_; p += 256) {
        float v = s[p];
        int rank = 0;
        for (int q = 0; q < NP_; ++q) {
            float u = s[q];
            rank += (u > v) || (u == v && q < p);
        }
        if (rank < KP_) topk[b * KP_ + rank] = p;
    }
}

// ---------------------------------------------------------------------------
// GEMM2: out[b,s,d] = sum_{k<256} acts[b,s,topk[b,k]] * PO[topk[b,k], d]
// One wave32 per 32x64 output macro-tile; K loop gathers through top-k indices.
// ---------------------------------------------------------------------------
__global__ __launch_bounds__(256) void k_gemm2(const float* __restrict__ acts,
                                               const float* __restrict__ PO,
                                               const int*   __restrict__ topk,
                                               float*       __restrict__ out) {
    const int lane = threadIdx.x & 31;
    const int wave = (blockIdx.x * blockDim.x + threadIdx.x) >> 5;
    const int ntilesN = DM_ / TILE_N;                       // 16
    const int tilesPerBatch = (S_ / TILE_M) * ntilesN;      // 1024
    const int b  = wave / tilesPerBatch;
    const int t  = wave % tilesPerBatch;
    const int m0 = (t / ntilesN) * TILE_M;
    const int n0 = (t % ntilesN) * TILE_N;

    const int mn = lane & 15;
    const int kb = (lane >> 4) * 2;

    const int*   idxb  = topk + b * KP_;
    const float* Arow0 = acts + (size_t)(b * S_ + m0 + mn) * NP_;
    const float* Arow1 = Arow0 + (size_t)16 * NP_;
    const float* POc   = PO + n0 + mn;                      // column base, row-indexed

    v8f c[MT_ * NT_];
#pragma unroll
    for (int i = 0; i < MT_ * NT_; ++i)
        c[i] = (v8f){0.f, 0.f, 0.f, 0.f, 0.f, 0.f, 0.f, 0.f};

#pragma unroll 2
    for (int k0 = 0; k0 < KP_; k0 += 4) {
        const int i0 = idxb[k0 + kb];
        const int i1 = idxb[k0 + kb + 1];
        v2f a0, a1;
        a0.x = Arow0[i0];  a0.y = Arow0[i1];     // gather selected acts columns
        a1.x = Arow1[i0];  a1.y = Arow1[i1];
#pragma unroll
        for (int j = 0; j < NT_; ++j) {
            v2f bm;
            bm.x = POc[(size_t)i0 * DM_ + j * 16];   // gather selected output rows
            bm.y = POc[(size_t)i1 * DM_ + j * 16];
            c[j * MT_ + 0] = __builtin_amdgcn_wmma_f32_16x16x4_f32(
                false, a0, false, bm, (short)0, c[j * MT_ + 0], false, false);
            c[j * MT_ + 1] = __builtin_amdgcn_wmma_f32_16x16x4_f32(
                false, a1, false, bm, (short)0, c[j * MT_ + 1], false, false);
        }
    }

    const int rbase = (lane >> 4) * 8;
#pragma unroll
    for (int j = 0; j < NT_; ++j) {
#pragma unroll
        for (int mi = 0; mi < MT_; ++mi) {
            float* Crow = out + (size_t)(b * S_ + m0 + mi * 16) * DM_ + n0 + j * 16 + mn;
            v8f cc = c[j * MT_ + mi];
#pragma unroll
            for (int r = 0; r < 8; ++r)
                Crow[(size_t)(rbase + r) * DM_] = cc[r];
        }
    }
}

// ---------------------------------------------------------------------------
extern "C" void kernel_launch(void* const* d_in, const int* in_sizes, int n_in,
                              void* d_out, int out_size, void* d_ws, size_t ws_size,
                              hipStream_t stream) {
    const float* A   = (const float*)d_in[0];   // [8,2048,128]  f32
    const float* PW  = (const float*)d_in[1];   // [1024,256]    f32
    const float* PO  = (const float*)d_in[2];   // [1024,1024]   f32
    const int*   IDX = (const int*)  d_in[3];   // [8,128]       i32
    float*       out = (float*)d_out;           // [8,2048,1024] f32

    char*  ws     = (char*)d_ws;
    float* acts   = (float*)ws;                                           // 64 MB
    float* scores = (float*)(ws + sizeof(float) * (size_t)B_ * S_ * NP_); // 32 KB
    int*   topk   = (int*)  (ws + sizeof(float) * ((size_t)B_ * S_ * NP_ + B_ * NP_));

    k_zero<<<(B_ * NP_ + 255) / 256, 256, 0, stream>>>(scores);

    const int waves1 = B_ * (S_ / TILE_M) * (NP_ / TILE_N);   // 8192 waves
    k_gemm1<<<waves1 / 8, 256, 0, stream>>>(A, PW, IDX, acts, scores);

    k_topk<<<B_, 256, 0, stream>>>(scores, topk);

    const int waves2 = B_ * (S_ / TILE_M) * (DM_ / TILE_N);   // 8192 waves
    k_gemm2<<<waves2 / 8, 256, 0, stream>>>(acts, PO, topk, out);
}